// GraphAttentionLayer_36206574305349
// MI455X (gfx1250) — compile-verified
//
#include <hip/hip_runtime.h>
#include <hip/hip_bf16.h>

typedef __attribute__((ext_vector_type(2))) float v2f;
typedef __attribute__((ext_vector_type(8))) float v8f;

#define GAT_B 4
#define GAT_T 8
#define GAT_N 1024
#define GAT_F 64
#define PSTRIDE 1026   // 16*1026*4B rows; stride%64==2 -> A-loads hit distinct LDS banks
#define NEG_INF_F (-1.0e12f)

// ---------------------------------------------------------------------------
// Kernel 1: h[r][o] = sum_k inp[r][k] * W[k][o], rows r over B*T*N = 32768.
// One block = 16 rows; 4 waves; wave w computes the 16x16 tile at cols w*16.
// fp32 WMMA 16x16x4, K-loop of 16.
// ---------------------------------------------------------------------------
__global__ __launch_bounds__(128) void gat_gemm_h(const float* __restrict__ inp,
                                                  const float* __restrict__ W,
                                                  float* __restrict__ h) {
  const int rowbase = blockIdx.x * 16;
  const int wave = threadIdx.x >> 5;
  const int lane = threadIdx.x & 31;
  const int m    = lane & 15;      // A: row M / B,D: col N
  const int half = lane >> 4;      // 0: K=0,1   1: K=2,3
  const int kl   = half * 2;
  const int o0   = wave * 16;

  const float* arow = inp + (size_t)(rowbase + m) * GAT_F;
  v8f acc = {};
#pragma unroll
  for (int kk = 0; kk < 16; ++kk) {
    const int k0 = kk * 4;
    v2f a = *(const v2f*)(arow + k0 + kl);            // 8B-aligned pair
    v2f b;
    b.x = W[(k0 + kl) * GAT_F + o0 + m];
    b.y = W[(k0 + kl + 1) * GAT_F + o0 + m];
    acc = __builtin_amdgcn_wmma_f32_16x16x4_f32(false, a, false, b,
                                                (short)0, acc, false, false);
  }
  float* orow = h + (size_t)rowbase * GAT_F + o0 + m;
#pragma unroll
  for (int v = 0; v < 8; ++v) {
    const int r = v + half * 8;                        // C/D layout: M = v (+8 for upper half)
    orow[(size_t)r * GAT_F] = acc[v];
  }
}

// ---------------------------------------------------------------------------
// Kernel 2: s1[r] = h[r]·a[0:64],  s2[r] = h[r]·a[64:128]
// ---------------------------------------------------------------------------
__global__ __launch_bounds__(256) void gat_scores(const float* __restrict__ h,
                                                  const float* __restrict__ a,
                                                  float* __restrict__ s1,
                                                  float* __restrict__ s2) {
  const int r = blockIdx.x * blockDim.x + threadIdx.x;
  if (r >= GAT_B * GAT_T * GAT_N) return;
  const float4* hr = (const float4*)(h + (size_t)r * GAT_F);
  float acc1 = 0.f, acc2 = 0.f;
#pragma unroll
  for (int q = 0; q < 16; ++q) {
    const float4 hv = hr[q];
    acc1 += hv.x * a[4 * q]      + hv.y * a[4 * q + 1]
          + hv.z * a[4 * q + 2]  + hv.w * a[4 * q + 3];
    acc2 += hv.x * a[64 + 4 * q]     + hv.y * a[64 + 4 * q + 1]
          + hv.z * a[64 + 4 * q + 2] + hv.w * a[64 + 4 * q + 3];
  }
  s1[r] = acc1;
  s2[r] = acc2;
}

// ---------------------------------------------------------------------------
// Kernel 3: fused attention block. One block per (b,t, 16-row tile):
//   P[i][j] = adj>0 ? lrelu(s1_i+s2_j)*adj : -1e12      (in LDS)
//   stable row softmax; fold m_j into P, (m_i / sum_i) into a row scale
//   D = P @ H  via fp32 WMMA (H staged through LDS in 64-row chunks)
//   out = elu(scale_i * D)
// ---------------------------------------------------------------------------
__global__ __launch_bounds__(128) void gat_attn(const float* __restrict__ adj,
                                                const float* __restrict__ att_mask,
                                                const float* __restrict__ h,
                                                const float* __restrict__ s1,
                                                const float* __restrict__ s2,
                                                float* __restrict__ out) {
  __shared__ float P[16 * PSTRIDE];
  __shared__ __attribute__((aligned(16))) float Hs[GAT_F * GAT_F];
  __shared__ float s1s[16];
  __shared__ float red[16 * 8];
  __shared__ float rowmax[16];
  __shared__ float rowscale[16];

  const int bt   = blockIdx.x >> 6;      // 0..31
  const int tile = blockIdx.x & 63;      // 0..63
  const int b    = bt >> 3;
  const int t    = bt & 7;
  const int i0   = tile * 16;            // node-row base
  const int tid  = threadIdx.x;
  const size_t base_bt = (size_t)bt * GAT_N;

  if (tid < 16) s1s[tid] = s1[base_bt + i0 + tid];
  __syncthreads();

  // ---- raw masked scores into LDS ----
  const float* adjbase = adj + ((size_t)t * GAT_N + i0) * GAT_N;
  for (int idx = tid; idx < 16 * GAT_N; idx += 128) {
    const int i = idx >> 10;
    const int j = idx & (GAT_N - 1);
    float e = s1s[i] + s2[base_bt + j];
    e = e > 0.f ? e : 0.2f * e;
    const float ad = adjbase[(size_t)i * GAT_N + j];
    P[i * PSTRIDE + j] = (ad > 0.f) ? e * ad : NEG_INF_F;
  }
  __syncthreads();

  // ---- row max (16 rows x 8 threads each) ----
  {
    const int i = tid >> 3, l = tid & 7;
    float mx = -3.0e38f;
    for (int j = l; j < GAT_N; j += 8) mx = fmaxf(mx, P[i * PSTRIDE + j]);
    red[i * 8 + l] = mx;
  }
  __syncthreads();
  if (tid < 16) {
    float mx = red[tid * 8];
#pragma unroll
    for (int l = 1; l < 8; ++l) mx = fmaxf(mx, red[tid * 8 + l]);
    rowmax[tid] = mx;
  }
  __syncthreads();

  // ---- exp pass: sum unmasked, store p * m_j ----
  {
    const int i = tid >> 3, l = tid & 7;
    const float mx = rowmax[i];
    float sum = 0.f;
    for (int j = l; j < GAT_N; j += 8) {
      const float p = __expf(P[i * PSTRIDE + j] - mx);
      sum += p;
      const float mj = att_mask[((size_t)b * GAT_N + j) * GAT_T + t];
      P[i * PSTRIDE + j] = p * mj;
    }
    red[i * 8 + l] = sum;
  }
  __syncthreads();
  if (tid < 16) {
    float sum = 0.f;
#pragma unroll
    for (int l = 0; l < 8; ++l) sum += red[tid * 8 + l];
    const float mi = att_mask[((size_t)b * GAT_N + i0 + tid) * GAT_T + t];
    rowscale[tid] = mi / sum;
  }
  __syncthreads();

  // ---- D = P @ H via fp32 WMMA, H chunked through LDS ----
  const int wave = tid >> 5;
  const int lane = tid & 31;
  const int m    = lane & 15;
  const int half = lane >> 4;
  const int kl   = half * 2;
  const int o0   = wave * 16;
  const float* hbt = h + base_bt * GAT_F;

  v8f acc = {};
  for (int kb = 0; kb < GAT_N; kb += GAT_F) {
    __syncthreads();                                   // protect Hs reuse
    const float4* src = (const float4*)(hbt + (size_t)kb * GAT_F);
#pragma unroll
    for (int c = 0; c < (GAT_F * GAT_F / 4) / 128; ++c) {
      const int idx = tid + c * 128;
      ((float4*)Hs)[idx] = src[idx];
    }
    __syncthreads();
#pragma unroll
    for (int kk = 0; kk < 16; ++kk) {
      const int k0 = kk * 4;
      v2f a = *(const v2f*)&P[m * PSTRIDE + kb + k0 + kl];
      v2f bb;
      bb.x = Hs[(k0 + kl) * GAT_F + o0 + m];
      bb.y = Hs[(k0 + kl + 1) * GAT_F + o0 + m];
      acc = __builtin_amdgcn_wmma_f32_16x16x4_f32(false, a, false, bb,
                                                  (short)0, acc, false, false);
    }
  }

  // ---- scale + ELU + store ----
  float* outbase = out + (base_bt + i0) * GAT_F;
#pragma unroll
  for (int v = 0; v < 8; ++v) {
    const int r = v + half * 8;
    float val = acc[v] * rowscale[r];
    val = val > 0.f ? val : (__expf(val) - 1.f);
    outbase[(size_t)r * GAT_F + o0 + m] = val;
  }
}

// ---------------------------------------------------------------------------
extern "C" void kernel_launch(void* const* d_in, const int* in_sizes, int n_in,
                              void* d_out, int out_size, void* d_ws, size_t ws_size,
                              hipStream_t stream) {
  const float* adj      = (const float*)d_in[0];  // (T,N,N)
  const float* inp      = (const float*)d_in[1];  // (B,T,N,FI)
  const float* att_mask = (const float*)d_in[2];  // (B,N,T)
  const float* W        = (const float*)d_in[3];  // (FI,FO)
  const float* a        = (const float*)d_in[4];  // (2*FO,1)
  float* out = (float*)d_out;

  const int nrows = GAT_B * GAT_T * GAT_N;        // 32768
  float* h  = (float*)d_ws;                        // nrows*64 floats (8 MB)
  float* s1 = h + (size_t)nrows * GAT_F;           // 32768 floats
  float* s2 = s1 + nrows;                          // 32768 floats

  gat_gemm_h<<<nrows / 16, 128, 0, stream>>>(inp, W, h);
  gat_scores<<<(nrows + 255) / 256, 256, 0, stream>>>(h, a, s1, s2);
  gat_attn<<<GAT_B * GAT_T * (GAT_N / 16), 128, 0, stream>>>(adj, att_mask, h, s1, s2, out);
}